// GeoGCN_44306882625628
// MI455X (gfx1250) — compile-verified
//
#include <hip/hip_runtime.h>

// GCN forward for MI455X (gfx1250, wave32).
// GEMMs use V_WMMA_F32_16X16X4_F32 (exact fp32). Edge aggregation uses
// float4 gathers + hardware f32 atomics (L2-resident: 12.8MB << 192MB L2).

typedef __attribute__((ext_vector_type(2))) float v2f;
typedef __attribute__((ext_vector_type(4))) float v4f;
typedef __attribute__((ext_vector_type(8))) float v8f;

#define N_NODES 50000
#define N_EDGES 800000
#define IN_C    16
#define HID_C   64
#define OUT_C   12
#define N_TILES (N_NODES / 16)   // 3125 exactly

__device__ __forceinline__ v8f wmma_k4(v2f a, v2f b, v8f c) {
    // (neg_a, A, neg_b, B, c_mod, C, reuse_a, reuse_b)
    return __builtin_amdgcn_wmma_f32_16x16x4_f32(false, a, false, b, (short)0, c,
                                                 false, false);
}

// Hardware GLOBAL_ATOMIC_ADD_F32 (non-returning, STOREcnt) — avoids CAS-loop
// expansion that plain atomicAdd(float*) can produce for fine-grained safety.
__device__ __forceinline__ void atom_add_f32(float* p, float v) {
    unsafeAtomicAdd(p, v);
}

// ---------------- degree / normalization precompute ----------------
__global__ void k_deg_init(float* deg) {
    int v = blockIdx.x * 256 + threadIdx.x;
    if (v < N_NODES) deg[v] = 1.0f;  // self-loop weight
}

__global__ void k_deg_scatter(const long long* __restrict__ ei,
                              const float* __restrict__ ew, float* deg) {
    int e = blockIdx.x * 256 + threadIdx.x;
    if (e < N_EDGES) atom_add_f32(deg + ei[N_EDGES + e], ew[e]);
}

__global__ void k_dinv(const float* __restrict__ deg, float* __restrict__ dinv) {
    int v = blockIdx.x * 256 + threadIdx.x;
    if (v < N_NODES) {
        float d = deg[v];
        dinv[v] = d > 0.f ? rsqrtf(d) : 0.f;
    }
}

__global__ void k_norm(const long long* __restrict__ ei, const float* __restrict__ ew,
                       const float* __restrict__ dinv, float* __restrict__ nrm) {
    int e = blockIdx.x * 256 + threadIdx.x;
    if (e < N_EDGES)
        nrm[e] = dinv[ei[e]] * ew[e] * dinv[ei[N_EDGES + e]];
}

// ---------------- input projection: H = relu(x @ W_in + b_in) ----------------
__global__ void k_input_proj(const float* __restrict__ x, const float* __restrict__ Win,
                             const float* __restrict__ bin, float* __restrict__ H) {
    __shared__ __align__(16) float WT[HID_C * 18];  // WT[n][k], stride 18 (8B-aligned rows)
    int tid = threadIdx.x;
    for (int i = tid; i < IN_C * HID_C; i += 256) {
        int k = i / HID_C, n = i % HID_C;   // Win[k][n]
        WT[n * 18 + k] = Win[i];
    }
    __syncthreads();
    int wave = tid >> 5, lane = tid & 31;
    int tile = blockIdx.x * 8 + wave;
    if (tile >= N_TILES) return;            // wave-uniform: EXEC all-ones for WMMA
    int r0 = tile * 16;
    int half = lane >> 4, nm = lane & 15;
    v8f acc[4] = {};
    for (int k0 = 0; k0 < IN_C; k0 += 4) {
        int k = k0 + half * 2;              // A layout: lane L holds K = 2*(L/16)+v
        v2f a = *(const v2f*)(x + (r0 + nm) * IN_C + k);
#pragma unroll
        for (int nt = 0; nt < 4; ++nt) {
            v2f b = *(const v2f*)(&WT[(nt * 16 + nm) * 18 + k]);
            acc[nt] = wmma_k4(a, b, acc[nt]);
        }
    }
#pragma unroll
    for (int nt = 0; nt < 4; ++nt)
#pragma unroll
        for (int v = 0; v < 8; ++v) {
            int row = r0 + v + 8 * half;    // C layout: VGPR v -> M = v + 8*(L/16)
            int col = nt * 16 + nm;
            float d = acc[nt][v] + bin[col];
            H[row * HID_C + col] = d > 0.f ? d : 0.f;
        }
}

// -------- conv GEMM: HW = Hin @ W ; AGG init = HW * (self-loop norm) --------
__global__ void k_gemm64(const float* __restrict__ Hin, const float* __restrict__ W,
                         const float* __restrict__ dinv,
                         float* __restrict__ HW, float* __restrict__ AGG) {
    __shared__ __align__(16) float WT[HID_C * 66];  // W transposed: WT[n][k]
    int tid = threadIdx.x;
    for (int i = tid; i < HID_C * HID_C; i += 256) {
        int k = i >> 6, n = i & 63;         // W[k][n]
        WT[n * 66 + k] = W[i];
    }
    __syncthreads();
    int wave = tid >> 5, lane = tid & 31;
    int tile = blockIdx.x * 8 + wave;
    if (tile >= N_TILES) return;
    int r0 = tile * 16;
    int half = lane >> 4, nm = lane & 15;
    v8f acc[4] = {};
    for (int k0 = 0; k0 < HID_C; k0 += 4) {
        int k = k0 + half * 2;
        v2f a = *(const v2f*)(Hin + (r0 + nm) * HID_C + k);
#pragma unroll
        for (int nt = 0; nt < 4; ++nt) {
            v2f b = *(const v2f*)(&WT[(nt * 16 + nm) * 66 + k]);
            acc[nt] = wmma_k4(a, b, acc[nt]);
        }
    }
#pragma unroll
    for (int v = 0; v < 8; ++v) {
        int row = r0 + v + 8 * half;
        float dv = dinv[row];
        float sn = dv * dv;                 // self-loop norm = 1/deg
#pragma unroll
        for (int nt = 0; nt < 4; ++nt) {
            int col = nt * 16 + nm;
            float d = acc[nt][v];
            HW[row * HID_C + col]  = d;
            AGG[row * HID_C + col] = d * sn;
        }
    }
}

// -------- edge scatter: AGG[dst] += norm * HW[src], 64 channels --------
__global__ void k_scatter64(const long long* __restrict__ ei, const float* __restrict__ nrm,
                            const float* __restrict__ HW, float* __restrict__ AGG) {
    int t = blockIdx.x * 256 + threadIdx.x;   // E*16 threads, 4 channels each
    int e = t >> 4, q = t & 15;
    if (e >= N_EDGES) return;
    long long s = ei[e], d = ei[N_EDGES + e];
    float w = nrm[e];
    v4f hv = *(const v4f*)(HW + s * HID_C + q * 4);  // 128-bit gather (L2 hit)
    float* base = AGG + d * HID_C + q * 4;
    atom_add_f32(base + 0, w * hv[0]);
    atom_add_f32(base + 1, w * hv[1]);
    atom_add_f32(base + 2, w * hv[2]);
    atom_add_f32(base + 3, w * hv[3]);
}

// ---- fused epilogue: AGG = relu((AGG + conv_b)*bn_g*invstd + bn_b) + Hres ----
__global__ void k_post(float* __restrict__ AGG, const float* __restrict__ Hres,
                       const float* __restrict__ cb, const float* __restrict__ bg,
                       const float* __restrict__ bb) {
    int t = blockIdx.x * 256 + threadIdx.x;   // N*16 float4 chunks
    if (t >= N_NODES * 16) return;
    const float invstd = rsqrtf(1.0f + 1e-5f);
    int c0 = (t & 15) * 4;
    v4f a = ((const v4f*)AGG)[t];
    v4f r = ((const v4f*)Hres)[t];
#pragma unroll
    for (int j = 0; j < 4; ++j) {
        int c = c0 + j;
        float val = (a[j] + cb[c]) * (bg[c] * invstd) + bb[c];
        a[j] = (val > 0.f ? val : 0.f) + r[j];
    }
    ((v4f*)AGG)[t] = a;
}

// -------- output GEMM: HW2 = H @ W_out (N padded 12->16); d_out init --------
__global__ void k_gemm_out(const float* __restrict__ Hin, const float* __restrict__ Wout,
                           const float* __restrict__ bout, const float* __restrict__ dinv,
                           float* __restrict__ HW2, float* __restrict__ out) {
    __shared__ __align__(16) float WT[16 * 66];
    int tid = threadIdx.x;
    for (int i = tid; i < 16 * HID_C; i += 256) {
        int n = i >> 6, k = i & 63;
        WT[n * 66 + k] = (n < OUT_C) ? Wout[k * OUT_C + n] : 0.f;
    }
    __syncthreads();
    int wave = tid >> 5, lane = tid & 31;
    int tile = blockIdx.x * 8 + wave;
    if (tile >= N_TILES) return;
    int r0 = tile * 16;
    int half = lane >> 4, nm = lane & 15;
    v8f acc = {};
    for (int k0 = 0; k0 < HID_C; k0 += 4) {
        int k = k0 + half * 2;
        v2f a = *(const v2f*)(Hin + (r0 + nm) * HID_C + k);
        v2f b = *(const v2f*)(&WT[nm * 66 + k]);
        acc = wmma_k4(a, b, acc);
    }
    if (nm < OUT_C) {                        // lane guard AFTER all WMMAs
#pragma unroll
        for (int v = 0; v < 8; ++v) {
            int row = r0 + v + 8 * half;
            float dv = dinv[row];
            float d = acc[v];
            HW2[row * OUT_C + nm] = d;
            out[row * OUT_C + nm] = d * dv * dv + bout[nm];  // self-loop + bias
        }
    }
}

// -------- edge scatter for output: out[dst] += norm * HW2[src], 12 ch --------
__global__ void k_scatter12(const long long* __restrict__ ei, const float* __restrict__ nrm,
                            const float* __restrict__ HW2, float* __restrict__ out) {
    int t = blockIdx.x * 256 + threadIdx.x;   // E*12 threads
    if (t >= N_EDGES * 12) return;
    int e = t / 12, c = t % 12;
    long long s = ei[e], d = ei[N_EDGES + e];
    atom_add_f32(out + d * OUT_C + c, nrm[e] * HW2[s * OUT_C + c]);
}

extern "C" void kernel_launch(void* const* d_in, const int* in_sizes, int n_in,
                              void* d_out, int out_size, void* d_ws, size_t ws_size,
                              hipStream_t stream) {
    const float*     x     = (const float*)d_in[0];
    const long long* ei    = (const long long*)d_in[1];   // int64 edge_index [2,E]
    const float*     ew    = (const float*)d_in[2];
    const float*     Win   = (const float*)d_in[3];
    const float*     bin   = (const float*)d_in[4];
    const float*     convw = (const float*)d_in[5];       // [2,64,64]
    const float*     convb = (const float*)d_in[6];       // [2,64]
    const float*     bng   = (const float*)d_in[7];
    const float*     bnb   = (const float*)d_in[8];
    const float*     Wout  = (const float*)d_in[9];
    const float*     bout  = (const float*)d_in[10];
    float* out = (float*)d_out;

    char* ws = (char*)d_ws;
    float* H    = (float*)(ws);              // 50000*64*4 = 12.8 MB
    float* AGG  = (float*)(ws + 12800000);   // 12.8 MB
    float* HW   = (float*)(ws + 25600000);   // 12.8 MB
    float* HW2  = (float*)(ws + 38400000);   //  2.4 MB
    float* deg  = (float*)(ws + 40800000);   //  0.2 MB
    float* dinv = (float*)(ws + 41000000);   //  0.2 MB
    float* nrm  = (float*)(ws + 41200000);   //  3.2 MB

    // degree / symmetric normalization (computed once, reused by all 3 convs)
    k_deg_init   <<<(N_NODES + 255) / 256, 256, 0, stream>>>(deg);
    k_deg_scatter<<<(N_EDGES + 255) / 256, 256, 0, stream>>>(ei, ew, deg);
    k_dinv       <<<(N_NODES + 255) / 256, 256, 0, stream>>>(deg, dinv);
    k_norm       <<<(N_EDGES + 255) / 256, 256, 0, stream>>>(ei, ew, dinv, nrm);

    // input projection
    k_input_proj<<<(N_TILES + 7) / 8, 256, 0, stream>>>(x, Win, bin, H);

    // conv layer 0: h in H -> result in AGG
    k_gemm64   <<<(N_TILES + 7) / 8, 256, 0, stream>>>(H, convw, dinv, HW, AGG);
    k_scatter64<<<N_EDGES * 16 / 256, 256, 0, stream>>>(ei, nrm, HW, AGG);
    k_post     <<<N_NODES * 16 / 256, 256, 0, stream>>>(AGG, H, convb, bng, bnb);

    // conv layer 1: h in AGG -> result in H (ping-pong)
    k_gemm64   <<<(N_TILES + 7) / 8, 256, 0, stream>>>(AGG, convw + HID_C * HID_C, dinv, HW, H);
    k_scatter64<<<N_EDGES * 16 / 256, 256, 0, stream>>>(ei, nrm, HW, H);
    k_post     <<<N_NODES * 16 / 256, 256, 0, stream>>>(H, AGG, convb + HID_C, bng + HID_C, bnb + HID_C);

    // output conv: h in H -> d_out
    k_gemm_out <<<(N_TILES + 7) / 8, 256, 0, stream>>>(H, Wout, bout, dinv, HW2, out);
    k_scatter12<<<N_EDGES * 12 / 256, 256, 0, stream>>>(ei, nrm, HW2, out);
}